// NAVQ_22333829939644
// MI455X (gfx1250) — compile-verified
//
#include <hip/hip_runtime.h>
#include <hip/hip_bf16.h>

typedef __attribute__((ext_vector_type(2))) float v2f;
typedef __attribute__((ext_vector_type(8))) float v8f;
typedef __attribute__((ext_vector_type(4))) unsigned int u32x4;
typedef __attribute__((ext_vector_type(8))) int i32x8;
typedef __attribute__((ext_vector_type(4))) int i32x4;

#define FEAT_DIM 128
#define KNN_K 15
#define E_MIN_F 0.34867844f   // float(0.9**10)

// ---------------- zero helper ----------------
__global__ void zero_f32(float* __restrict__ p, size_t n) {
  size_t i = (size_t)blockIdx.x * blockDim.x + threadIdx.x;
  size_t stride = (size_t)gridDim.x * blockDim.x;
  for (; i < n; i += stride) p[i] = 0.0f;
}

// ---------------- row squared norms ----------------
__global__ void row_norms(const float* __restrict__ v, float* __restrict__ out, int rows) {
  int i = blockIdx.x * blockDim.x + threadIdx.x;
  if (i >= rows) return;
  const float4* p = (const float4*)(v + (size_t)i * FEAT_DIM);
  float s = 0.0f;
#pragma unroll
  for (int t = 0; t < FEAT_DIM / 4; ++t) {
    float4 q = p[t];
    s += q.x * q.x + q.y * q.y + q.z * q.z + q.w * q.w;
  }
  out[i] = s;
}

// ---------------- TDM tile stage: global 2-D tile -> LDS ----------------
// rows x 128 f32 tile, contiguous rows (stride 128), LDS rows padded to 132
// floats via TDM pad: pad_interval=6 (every 128 DWORDs) pad_amount=3 (4 DWORDs).
__device__ __forceinline__ void tdm_load_tile(const float* gsrc, unsigned lds_addr,
                                              int tile_rows, int rows_avail) {
  unsigned long long ga = (unsigned long long)(uintptr_t)gsrc;

  u32x4 g0;
  g0[0] = 1u;                                                  // count=1, user descriptor
  g0[1] = lds_addr;                                            // LDS byte address
  g0[2] = (unsigned)(ga & 0xffffffffu);                        // global_addr[31:0]
  g0[3] = (unsigned)((ga >> 32) & 0x01ffffffu) | (2u << 30);   // global_addr[56:32], type=2

  unsigned td0 = (unsigned)FEAT_DIM;        // tensor_dim0
  unsigned td1 = (unsigned)rows_avail;      // tensor_dim1 (rows available from tile start)
  i32x8 g1;
  g1[0] = (int)((2u << 16) | (1u << 20) | (6u << 22) | (3u << 25)); // data_size=4B, pad on
  g1[1] = (int)((td0 & 0xffffu) << 16);                             // tensor_dim0[15:0] @ bits 63:48
  g1[2] = (int)((td0 >> 16) | ((td1 & 0xffffu) << 16));             // tensor_dim0[31:16], tensor_dim1[15:0]
  g1[3] = (int)((td1 >> 16) | ((unsigned)FEAT_DIM << 16));          // tensor_dim1[31:16], tile_dim0=128
  g1[4] = (int)(unsigned)tile_rows;                                 // tile_dim1, tile_dim2=0
  g1[5] = (int)(unsigned)FEAT_DIM;                                  // tensor_dim0_stride[31:0]
  g1[6] = 0;                                                        // stride hi, dim1_stride lo
  g1[7] = 0;

  i32x4 z4 = {0, 0, 0, 0};
  i32x8 z8 = {0, 0, 0, 0, 0, 0, 0, 0};
  __builtin_amdgcn_tensor_load_to_lds(g0, g1, z4, z4, z8, 0);
}

// ---------------- fp32 WMMA distance GEMM ----------------
// d2[i,j] = max(||x_i||^2 + ||c_j||^2 - 2 x_i . c_j, 0)
// Block = 256 threads (8 waves), tile 32(M) x 64(N), full K=128 staged in LDS
// by the Tensor Data Mover. Each wave owns one 16x16 C tile: 32 WMMAs.
__global__ __launch_bounds__(256) void d2_wmma(
    const float* __restrict__ x, const float* __restrict__ cvs,
    const float* __restrict__ xnorm, const float* __restrict__ cnorm,
    float* __restrict__ d2, int Bn, int Nn)
{
  __shared__ float As[32][FEAT_DIM + 4];   // +4 pad: stride 132 floats -> bank spread
  __shared__ float Bs[64][FEAT_DIM + 4];
  const int bm = blockIdx.x * 32;
  const int bn = blockIdx.y * 64;
  const int tid = threadIdx.x;

  if (tid == 0) {
    tdm_load_tile(x + (size_t)bm * FEAT_DIM,
                  (unsigned)(uintptr_t)(void*)&As[0][0], 32, Bn - bm);
    tdm_load_tile(cvs + (size_t)bn * FEAT_DIM,
                  (unsigned)(uintptr_t)(void*)&Bs[0][0], 64, Nn - bn);
    __builtin_amdgcn_s_wait_tensorcnt(0);
  }
  __syncthreads();

  const int wave = tid >> 5;
  const int lane = tid & 31;
  const int wm   = (wave >> 2) << 4;  // 0 or 16
  const int wn   = (wave & 3) << 4;   // 0,16,32,48
  const int mr   = lane & 15;         // A row / B col within tile
  const int half = lane >> 4;         // K sub-pair select

  v8f acc = {};
#pragma unroll
  for (int k = 0; k < FEAT_DIM; k += 4) {
    const int koff = k + half * 2;
    v2f a, b;
    a.x = As[wm + mr][koff];
    a.y = As[wm + mr][koff + 1];
    b.x = Bs[wn + mr][koff];
    b.y = Bs[wn + mr][koff + 1];
    acc = __builtin_amdgcn_wmma_f32_16x16x4_f32(
        /*neg_a=*/false, a, /*neg_b=*/false, b,
        /*c_mod=*/(short)0, acc, /*reuse_a=*/false, /*reuse_b=*/false);
  }

  // C/D layout: VGPR r -> row (r + 8*half), col (lane&15)
  const int col = bn + wn + mr;
  const float cn = cnorm[col];
#pragma unroll
  for (int r = 0; r < 8; ++r) {
    int row = bm + wm + half * 8 + r;
    float v = xnorm[row] + cn - 2.0f * acc[r];
    d2[(size_t)row * Nn + col] = fmaxf(v, 0.0f);
  }
}

// ---------------- top-15 per row + scatter (visits, cce) ----------------
// One wave per batch row. Each lane keeps a sorted local top-15 over its 128
// strided columns, then 15 rounds of cross-lane argmin-pop (lower index wins ties).
__global__ __launch_bounds__(256) void topk_scatter(
    const float* __restrict__ d2, float* __restrict__ visits,
    float* __restrict__ cce, int Bn, int Nn)
{
  const int wid  = (blockIdx.x * blockDim.x + threadIdx.x) >> 5;
  const int lane = threadIdx.x & 31;
  if (wid >= Bn) return;
  const float* row = d2 + (size_t)wid * Nn;

  float bv[KNN_K];
  int   bi[KNN_K];
#pragma unroll
  for (int t = 0; t < KNN_K; ++t) { bv[t] = 3.402823466e38f; bi[t] = 0x7fffffff; }

  for (int t = 0; t < Nn / 32; ++t) {
    int j = lane + t * 32;             // coalesced
    float v = row[j];
    if (v < bv[KNN_K - 1]) {           // strict < : earlier index wins ties
      int p = KNN_K - 1;
      while (p > 0 && v < bv[p - 1]) { bv[p] = bv[p - 1]; bi[p] = bi[p - 1]; --p; }
      bv[p] = v; bi[p] = j;
    }
  }

  int head = 0;
  int closest = -1;
  for (int s = 0; s < KNN_K; ++s) {
    float v  = (head < KNN_K) ? bv[head] : 3.402823466e38f;
    int   id = (head < KNN_K) ? bi[head] : 0x7fffffff;
    float rv = v; int ri = id;
    for (int off = 16; off > 0; off >>= 1) {
      float ov = __shfl_down(rv, off);
      int   oi = __shfl_down(ri, off);
      if (ov < rv || (ov == rv && oi < ri)) { rv = ov; ri = oi; }
    }
    rv = __shfl(rv, 0);
    ri = __shfl(ri, 0);
    if (ri == id && head < KNN_K) head++;   // owning lane pops (col indices unique per lane)
    if (s == 0) closest = ri;               // wave-uniform
    if (lane == 0)
      atomicAdd(&cce[(size_t)closest * Nn + ri], 1.0f);
  }
  if (lane == 0) atomicAdd(&visits[closest], 1.0f);
}

// ---------------- neighbor mask ----------------
__global__ void mask_kernel(const float* __restrict__ edges,
                            const float* __restrict__ conn,
                            const float* __restrict__ cce,
                            const float* __restrict__ visits,
                            unsigned char* __restrict__ mask, int Nn)
{
  size_t idx = (size_t)blockIdx.x * blockDim.x + threadIdx.x;
  size_t total = (size_t)Nn * Nn;
  if (idx >= total) return;
  int r = (int)(idx / (size_t)Nn);
  float c   = cce[idx];
  float enc = fmaxf(visits[r] - c, 0.0f);
  float base = fmaxf(edges[idx], (c > 0.0f) ? 1.0f : 0.0f);
  float val = base * powf(0.9f, enc);
  if (val < E_MIN_F) val = 0.0f;
  mask[idx] = (val > 0.0f && conn[idx] < 1.0f) ? (unsigned char)1 : (unsigned char)0;
}

// ---------------- per-row loss ----------------
// One wave per batch row: S1 = sum e_neg, S2 = sum e_neg*d_neg -> sum(w*d) = S2/S1.
__global__ __launch_bounds__(256) void loss_kernel(
    const float* __restrict__ x, const float* __restrict__ cvs,
    const int* __restrict__ labels, const float* __restrict__ d2,
    const unsigned char* __restrict__ mask, float* __restrict__ out,
    int Bn, int Nn)
{
  const int wid  = (blockIdx.x * blockDim.x + threadIdx.x) >> 5;
  const int lane = threadIdx.x & 31;
  if (wid >= Bn) return;
  const int lab = labels[wid];
  const float* drow = d2 + (size_t)wid * Nn;
  const unsigned char* mrow = mask + (size_t)lab * Nn;

  float s1 = 0.0f, s2 = 0.0f;
  for (int t = lane; t < Nn; t += 32) {
    float dn = drow[t] * (mrow[t] ? 1.0f : 0.0f);
    if (dn > 0.0f) {
      float e = expf(-0.001f * dn);
      s1 += e;
      s2 += e * dn;
    }
  }
  float dp = 0.0f;
  const float* xr = x + (size_t)wid * FEAT_DIM;
  const float* cr = cvs + (size_t)lab * FEAT_DIM;
  for (int t = lane; t < FEAT_DIM; t += 32) {
    float df = xr[t] - cr[t];
    dp += df * df;
  }
  for (int off = 16; off > 0; off >>= 1) {
    s1 += __shfl_down(s1, off);
    s2 += __shfl_down(s2, off);
    dp += __shfl_down(dp, off);
  }
  if (lane == 0) {
    float sub = (s1 > 0.0f) ? (s2 / s1) : 0.0f;   // NaN-w -> 0 case
    float mu = dp - sub;
    if (mu > 0.0f) atomicAdd(out, mu / (float)Bn);
  }
}

extern "C" void kernel_launch(void* const* d_in, const int* in_sizes, int n_in,
                              void* d_out, int out_size, void* d_ws, size_t ws_size,
                              hipStream_t stream) {
  const float* x      = (const float*)d_in[0];
  const float* cvs    = (const float*)d_in[1];
  const float* edges  = (const float*)d_in[2];
  const float* conn   = (const float*)d_in[3];
  const int*   labels = (const int*)d_in[4];
  float* out = (float*)d_out;

  const int Bn = in_sizes[0] / FEAT_DIM;   // 8192
  const int Nn = in_sizes[1] / FEAT_DIM;   // 4096

  // workspace layout
  float* ws     = (float*)d_ws;
  float* d2     = ws;                                    // Bn*Nn           (128 MB)
  float* cce    = d2 + (size_t)Bn * Nn;                  // Nn*Nn           ( 64 MB)
  float* visits = cce + (size_t)Nn * Nn;                 // Nn
  float* xnorm  = visits + Nn;                           // Bn
  float* cnorm  = xnorm + Bn;                            // Nn
  unsigned char* mask = (unsigned char*)(cnorm + Nn);    // Nn*Nn bytes     ( 16 MB)

  // 1) zero accumulators (cce+visits contiguous) and output
  size_t zn = (size_t)Nn * Nn + Nn;
  zero_f32<<<2048, 256, 0, stream>>>(cce, zn);
  zero_f32<<<1, 1, 0, stream>>>(out, 1);

  // 2) row norms
  row_norms<<<(Bn + 255) / 256, 256, 0, stream>>>(x, xnorm, Bn);
  row_norms<<<(Nn + 255) / 256, 256, 0, stream>>>(cvs, cnorm, Nn);

  // 3) fp32 WMMA distance matrix (tiles staged by TDM)
  dim3 ggrid(Bn / 32, Nn / 64);
  d2_wmma<<<ggrid, 256, 0, stream>>>(x, cvs, xnorm, cnorm, d2, Bn, Nn);

  // 4) top-15 + scatter into visits/cce (one wave per row)
  topk_scatter<<<(Bn * 32) / 256, 256, 0, stream>>>(d2, visits, cce, Bn, Nn);

  // 5) neighbor mask
  size_t total = (size_t)Nn * Nn;
  mask_kernel<<<(unsigned)((total + 255) / 256), 256, 0, stream>>>(edges, conn, cce, visits, mask, Nn);

  // 6) loss (one wave per row)
  loss_kernel<<<(Bn * 32) / 256, 256, 0, stream>>>(x, cvs, labels, d2, mask, out, Bn, Nn);
}